// LRUCache_8675833938082
// MI455X (gfx1250) — compile-verified
//
#include <hip/hip_runtime.h>
#include <stdint.h>

// ---- problem constants (match reference) ----
constexpr int H        = 8;
constexpr int ND       = 4096;   // NUM_DRAM_PAGES
constexpr int NH       = 819;    // NUM_HBM_PAGES = int(4096*0.2)
constexpr int TOPK     = 256;
constexpr int PSZ      = 16;     // page_size
constexpr int HD       = 128;    // head_dim
constexpr int PAGE_ELEMS = PSZ * 2 * HD;   // 4096 floats / page
constexpr int PAGE_VEC   = PAGE_ELEMS / 4; // 1024 float4 / page
constexpr long long OUT_PER = (long long)H * NH * PSZ * HD; // floats per output tensor

typedef float __attribute__((ext_vector_type(4))) f4;
typedef int   __attribute__((ext_vector_type(4))) v4i;

#define AS1 __attribute__((address_space(1)))
#define AS3 __attribute__((address_space(3)))

#if __has_builtin(__builtin_amdgcn_global_load_async_to_lds_b128)
#define HAVE_ASYNC_LOAD 1
#endif
#if __has_builtin(__builtin_amdgcn_global_store_async_from_lds_b128)
#define HAVE_ASYNC_STORE 1
#endif

// int->pointer casts are always legal; global AS shares flat VA values, and
// AS3 pointers are 32-bit so the flat shared address truncates to the LDS
// byte offset.
__device__ __forceinline__ AS1 v4i* as_global_v4i(const void* p) {
  return (AS1 v4i*)(uintptr_t)p;
}
__device__ __forceinline__ AS3 v4i* as_lds_v4i(const void* p) {
  return (AS3 v4i*)(uintptr_t)p;
}

__device__ __forceinline__ void wait_asynccnt0() {
#if __has_builtin(__builtin_amdgcn_s_wait_asynccnt)
  __builtin_amdgcn_s_wait_asynccnt(0);
#else
  asm volatile("s_wait_asynccnt 0" ::: "memory");
#endif
}

// -------------------------------------------------------------------------
// Kernel 1: per-head LRU control. One block per head, 256 threads.
// Computes src_page[h*NH + slot] = dram page to load into that slot (or -1
// meaning "keep current hbm contents"). Inputs are NOT mutated.
// -------------------------------------------------------------------------
__global__ __launch_bounds__(256) void lru_control(
    const float* __restrict__ pat_g,   // (H, NH)
    const int*   __restrict__ d2h_g,   // (H, ND)
    const int*   __restrict__ topk_g,  // (H, TOPK)
    const int*   __restrict__ step_g,  // (1,)
    int*         __restrict__ src_page)// (H, NH) scratch
{
  __shared__ float pat_s[NH];
  __shared__ int   order_s[NH];
  __shared__ int   topk_s[TOPK];
  __shared__ int   slot_s[TOPK];

  const int h   = blockIdx.x;
  const int tid = threadIdx.x;
  const float step_f = (float)(step_g[0] + 1);

  for (int i = tid; i < NH; i += 256) pat_s[i] = pat_g[h * NH + i];
  {
    int t = topk_g[h * TOPK + tid];
    topk_s[tid] = t;
    slot_s[tid] = d2h_g[h * ND + t];
  }
  __syncthreads();

  // Hits refresh their slot's timestamp (slots unique: topk unique, d2h injective).
  if (slot_s[tid] >= 0) pat_s[slot_s[tid]] = step_f;
  __syncthreads();

  // Stable argsort via rank counting: rank(i) = #{p[j]<p[i]} + #{j<i, p[j]==p[i]}.
  for (int i = tid; i < NH; i += 256) {
    const float pi = pat_s[i];
    int r = 0;
    for (int j = 0; j < NH; ++j) {
      const float pj = pat_s[j];
      r += (int)((pj < pi) || ((pj == pi) && (j < i)));
    }
    order_s[r] = i;  // order_s = argsort(pat_s), stable
  }
  for (int i = tid; i < NH; i += 256) src_page[h * NH + i] = -1;
  __syncthreads();

  // Misses: rank = prefix count of misses; victim = order[rank] (LRU slots).
  // Hit slots carry timestamp step_f (maximum) so they sort last and are
  // never victims (misses <= 256 << NH).
  if (slot_s[tid] < 0) {
    int r = 0;
    for (int j = 0; j < tid; ++j) r += (int)(slot_s[j] < 0);
    const int victim = order_s[r];
    src_page[h * NH + victim] = topk_s[tid];  // victims unique -> race-free
  }
}

// -------------------------------------------------------------------------
// Kernel 2: page copy + K/V deinterleave. One block per (head, hbm slot).
// The deinterleave is a pure destination-address permutation (thread v's
// output element is smem[v], which it loaded itself), so the whole 16KB page
// flows global -> LDS -> global on the CDNA5 async datapath (ASYNCcnt) with
// B128 transfers, never touching data VGPRs.
// Page layout: float e at (t, c, d): e = t*256 + c*128 + d.
//   K out row (h*NH+s)*16 + t  <- c == 0 half
//   V out row (h*NH+s)*16 + t  <- c == 1 half
// -------------------------------------------------------------------------
__global__ __launch_bounds__(256) void lru_copy(
    const float* __restrict__ dram,     // (H, ND, PAGE_ELEMS)
    const float* __restrict__ hbm,      // (H, NH, PAGE_ELEMS)
    const int*   __restrict__ src_page, // (H, NH)
    float*       __restrict__ out)      // k (OUT_PER) then v (OUT_PER)
{
  const int page = blockIdx.x;          // 0 .. H*NH-1
  const int h    = page / NH;
  const int s    = page - h * NH;
  const int tid  = threadIdx.x;

  const int src = src_page[page];
  const f4* srcp = (src >= 0)
      ? (const f4*)(dram + ((size_t)h * ND + (size_t)src) * PAGE_ELEMS)
      : (const f4*)(hbm + (size_t)page * PAGE_ELEMS);

  f4* outk = (f4*)out;
  f4* outv = (f4*)out + (size_t)(OUT_PER / 4);
  const size_t rowbase = (size_t)page * PSZ;  // output row index base

#ifdef HAVE_ASYNC_LOAD
  __shared__ f4 smem[PAGE_VEC];  // 16 KB page staging buffer

  // Stage global -> LDS: each lane issues 4x GLOBAL_LOAD_ASYNC_TO_LDS_B128.
  for (int v = tid; v < PAGE_VEC; v += 256) {
    __builtin_amdgcn_global_load_async_to_lds_b128(
        as_global_v4i(srcp + v), as_lds_v4i(&smem[v]), 0, 0);
  }
  wait_asynccnt0();  // this wave's LDS data is ready; no cross-thread reads

  for (int v = tid; v < PAGE_VEC; v += 256) {
    const int t  = v >> 6;         // 64 float4 per t (256 floats)
    const int c  = (v >> 5) & 1;   // 32 float4 per c-half (128 floats)
    const int dv = v & 31;         // float4 index within head_dim
    const size_t o = (rowbase + (size_t)t) * (HD / 4) + (size_t)dv;
    f4* dst = (c == 0) ? (outk + o) : (outv + o);
#ifdef HAVE_ASYNC_STORE
    // LDS -> global, async B128; S_ENDPGM's implicit wait-idle drains ASYNCcnt.
    __builtin_amdgcn_global_store_async_from_lds_b128(
        as_global_v4i(dst), as_lds_v4i(&smem[v]), 0, 0);
#else
    *dst = smem[v];
#endif
  }
#else
  // Fallback: direct B128 register copy (no LDS needed: pure addr permutation).
  for (int v = tid; v < PAGE_VEC; v += 256) {
    const int t  = v >> 6;
    const int c  = (v >> 5) & 1;
    const int dv = v & 31;
    const size_t o = (rowbase + (size_t)t) * (HD / 4) + (size_t)dv;
    const f4 val = srcp[v];
    if (c == 0) outk[o] = val;
    else        outv[o] = val;
  }
#endif
}

extern "C" void kernel_launch(void* const* d_in, const int* in_sizes, int n_in,
                              void* d_out, int out_size, void* d_ws, size_t ws_size,
                              hipStream_t stream) {
  const float* dram = (const float*)d_in[0];  // dram_kv_cache (H, ND, PAGE_ELEMS)
  const float* hbm  = (const float*)d_in[1];  // hbm_kv_cache  (H, NH, PAGE_ELEMS)
  const float* pat  = (const float*)d_in[2];  // page_access_time (H, NH)
  const int*   d2h  = (const int*)d_in[3];    // d2h_map (H, ND)
  // d_in[4] = h2d_map: only affects un-returned maps, not k/v output.
  const int*   topk = (const int*)d_in[5];    // topk_idx (H, TOPK)
  const int*   step = (const int*)d_in[6];    // current_step (1,)

  int* src_page = (int*)d_ws;  // H*NH ints, rewritten fully each call

  lru_control<<<H, 256, 0, stream>>>(pat, d2h, topk, step, src_page);
  lru_copy<<<H * NH, 256, 0, stream>>>(dram, hbm, src_page, (float*)d_out);
}